// MacUnit_29008209117228
// MI455X (gfx1250) — compile-verified
//
#include <hip/hip_runtime.h>
#include <hip/hip_bf16.h>

#define LOG2E   1.4426950408889634f
#define INV_2PI 0.15915494309189535f

// sigmoid(x) = 0.5*tanh(x/2) + 0.5  -> single v_tanh_f32 on gfx1250.
__device__ __forceinline__ float fast_sigmoid(float d) {
#if __has_builtin(__builtin_amdgcn_tanhf)
    return fmaf(0.5f, __builtin_amdgcn_tanhf(0.5f * d), 0.5f);
#else
    float e = __builtin_amdgcn_exp2f(-d * LOG2E);
    return __builtin_amdgcn_rcpf(1.0f + e);
#endif
}

// idx = 5*sigmoid(d), fused: fma(2.5, tanh(d/2), 2.5) -> 1 TRANS + 2 VALU.
__device__ __forceinline__ float fast_idx5(float d) {
#if __has_builtin(__builtin_amdgcn_tanhf)
    return fmaf(2.5f, __builtin_amdgcn_tanhf(0.5f * d), 2.5f);
#else
    float e = __builtin_amdgcn_exp2f(-d * LOG2E);
    return 5.0f * __builtin_amdgcn_rcpf(1.0f + e);
#endif
}

// data:(B,512)  params:(1024,)  tables:(5,)  out:(B,1024)
// out[b,ch] = f(data[b,ch/2]) -- purely elementwise; TRANS-pipe bound.
__global__ __launch_bounds__(256) void macunit_kernel(
    const float* __restrict__ data,
    const float* __restrict__ angles,
    const float* __restrict__ velocity,
    const float* __restrict__ in_w,
    const float* __restrict__ in_b,
    const float* __restrict__ out_w,
    const float* __restrict__ out_b,
    float* __restrict__ out,
    int batch)
{
    // Interpolation pair table: entry p = {v[p], a'[p], v[p+1], a'[p+1]}, p in [0,3],
    // with a' = angle * (1/2pi) pre-scaled for v_sin_f32/v_cos_f32 (revolutions).
    // One ds_load_b128 fetches both endpoints of both tables.
    __shared__ float4 table[4];
    const int tid = threadIdx.x;
    if (tid < 4) {
        table[tid] = make_float4(velocity[tid],     angles[tid]     * INV_2PI,
                                 velocity[tid + 1], angles[tid + 1] * INV_2PI);
    }
    __syncthreads();

    const int lane_in_row = tid & 127;   // 128 threads cover one batch row
    const int row_in_blk  = tid >> 7;    // 2 rows per 256-thread block
    const int c0  = lane_in_row * 4;     // data column base (4 cols/thread)
    const int ch0 = c0 * 2;              // output channel base (8 ch/thread)

    // Per-channel parameters: loop-invariant, keep in VGPRs (B128 loads).
    const float4 iw0 = *(const float4*)(in_w  + ch0);
    const float4 iw1 = *(const float4*)(in_w  + ch0 + 4);
    const float4 ib0 = *(const float4*)(in_b  + ch0);
    const float4 ib1 = *(const float4*)(in_b  + ch0 + 4);
    const float4 ow0 = *(const float4*)(out_w + ch0);
    const float4 ow1 = *(const float4*)(out_w + ch0 + 4);
    const float4 ob0 = *(const float4*)(out_b + ch0);
    const float4 ob1 = *(const float4*)(out_b + ch0 + 4);
    const float ow[8] = {ow0.x, ow0.y, ow0.z, ow0.w, ow1.x, ow1.y, ow1.z, ow1.w};
    const float ob[8] = {ob0.x, ob0.y, ob0.z, ob0.w, ob1.x, ob1.y, ob1.z, ob1.w};

    const int groups = gridDim.x * 2;
    for (int row = blockIdx.x * 2 + row_in_blk; row < batch; row += groups) {
        // Prefetch next row's tile while this row's transcendentals run.
        const int nrow = row + groups;
        if (nrow < batch) {
            __builtin_prefetch(data + (size_t)nrow * 512 + c0, 0, 1); // global_prefetch_b8
        }

        const float4 d4 = *(const float4*)(data + (size_t)row * 512 + c0);

        // Expansion: ch = 2c+f, x = data[c]*in_w[ch] + in_b[ch]
        float x[8];
        x[0] = fmaf(d4.x, iw0.x, ib0.x);
        x[1] = fmaf(d4.x, iw0.y, ib0.y);
        x[2] = fmaf(d4.y, iw0.z, ib0.z);
        x[3] = fmaf(d4.y, iw0.w, ib0.w);
        x[4] = fmaf(d4.z, iw1.x, ib1.x);
        x[5] = fmaf(d4.z, iw1.y, ib1.y);
        x[6] = fmaf(d4.w, iw1.z, ib1.z);
        x[7] = fmaf(d4.w, iw1.w, ib1.w);

        // 5 refinement steps. The reference's clamp logic collapses to
        // p = clamp(floor(idx),0,3), endpoints (p, p+1); pos = idx - p
        // (extrapolates past 1 at the top edge, matching the clamped accessor).
        #pragma unroll
        for (int s = 0; s < 5; ++s) {
            #pragma unroll
            for (int j = 0; j < 8; ++j) {
                const float d   = x[j];
                const float idx = fast_idx5(d);           // 5*sigmoid(d), fused
                float pf = floorf(idx);
                pf = fminf(fmaxf(pf, 0.0f), 3.0f);
                const float4 t = table[(int)pf];          // ds_load_b128
                const float pos  = idx - pf;
                const float velo = fmaf(pos, t.z - t.x, t.x);
                const float ang  = fmaf(pos, t.w - t.y, t.y); // already in revolutions
                const float cs = __builtin_amdgcn_cosf(ang);  // v_cos_f32
                const float sn = __builtin_amdgcn_sinf(ang);  // v_sin_f32
                const float stp = velo * fmaf(d, sn, cs); // velo*cos + d*velo*sin
                x[j] = fmaf(stp, 0.2f, d);                // x += step/NUM_STEPS
            }
        }

        // Gated output: out = x * sigmoid(x*out_w + out_b); OUT_CF=SP_DIMS=1 so the
        // reference's final sum is an identity.
        float r[8];
        #pragma unroll
        for (int j = 0; j < 8; ++j) {
            r[j] = x[j] * fast_sigmoid(fmaf(x[j], ow[j], ob[j]));
        }
        float* op = out + (size_t)row * 1024 + ch0;
        *(float4*)(op)     = make_float4(r[0], r[1], r[2], r[3]);
        *(float4*)(op + 4) = make_float4(r[4], r[5], r[6], r[7]);
    }
}

extern "C" void kernel_launch(void* const* d_in, const int* in_sizes, int n_in,
                              void* d_out, int out_size, void* d_ws, size_t ws_size,
                              hipStream_t stream) {
    const float* data     = (const float*)d_in[0];
    const float* angles   = (const float*)d_in[1];
    const float* velocity = (const float*)d_in[2];
    const float* in_w     = (const float*)d_in[3];
    const float* in_b     = (const float*)d_in[4];
    const float* out_w    = (const float*)d_in[5];
    const float* out_b    = (const float*)d_in[6];
    float* out = (float*)d_out;

    const int batch = in_sizes[0] / 512;   // 16384

    int blocks = 2048;                     // 16384 waves; grid-stride over rows
    if (blocks * 2 > batch) blocks = (batch + 1) / 2;

    macunit_kernel<<<dim3(blocks), dim3(256), 0, stream>>>(
        data, angles, velocity, in_w, in_b, out_w, out_b, out, batch);
}